// LocalConvolution_51101520887822
// MI455X (gfx1250) — compile-verified
//
#include <hip/hip_runtime.h>

// Local (unshared-weight) convolution, K=7, pad=3, stride=1.
// x:      [4, 256, 64, 64]  f32
// weight: [4, 1, 16, 49, 64, 64] f32
// out:    [4, 256, 64, 64]  f32,  out channel c = g*16+cc uses weight channel cc=c%16.
//
// Memory-bound (AI ~4.8 F/B, ~85MB -> ~3.7us floor @ 23.3TB/s). No matmul
// structure exists (per-pixel weights), so the CDNA5-specific path used here is
// the async tensor/data-mover pipeline: global_load_async_to_lds_b32 with
// double buffering + s_wait_asynccnt, not WMMA (which would be ~5x slower).

#define KK7   7
#define PAD3  3
#define TW    32
#define THT   8
#define HALO_W (TW + KK7 - 1)          // 38
#define HALO_H (THT + KK7 - 1)         // 14
#define HALO_N (HALO_W * HALO_H)       // 532
#define WC    16
#define GN    16
#define CHN   256
#define IMG   64
#define PLANE (IMG * IMG)              // 4096

__device__ __forceinline__ void async_f32_to_lds(const float* saddr,
                                                 unsigned voff_bytes,
                                                 unsigned lds_byte_addr) {
  // GVS form: mem_addr = SADDR(64b sgpr pair) + VADDR(32b vgpr offset)
  asm volatile("global_load_async_to_lds_b32 %0, %1, %2"
               :: "v"(lds_byte_addr), "v"(voff_bytes), "s"(saddr)
               : "memory");
}

__device__ __forceinline__ void wait_asynccnt0() {
  asm volatile("s_wait_asynccnt 0x0" ::: "memory");
}

__global__ __launch_bounds__(256, 1)
void local_conv7x7_async_kernel(const float* __restrict__ xin,
                                const float* __restrict__ wgt,
                                float* __restrict__ out) {
  __shared__ float tile[2][HALO_N];

  const int tid = threadIdx.x;
  const int tx  = tid & 31;          // lane   -> x within tile
  const int ty  = tid >> 5;          // waveId -> y within tile
  const int x0  = (blockIdx.x & 1) * TW;
  const int y0  = (blockIdx.x >> 1) * THT;
  const int cc  = blockIdx.y;        // weight channel 0..15
  const int b   = blockIdx.z;
  const int xx  = x0 + tx;
  const int yy  = y0 + ty;

  // ---- stage 49 per-pixel weights in VGPRs (streamed once, NT) ----
  const float* wptr = wgt + (((size_t)b * WC + cc) * 49) * PLANE
                          + (size_t)yy * IMG + xx;
  float w[49];
#pragma unroll
  for (int k = 0; k < 49; ++k)
    w[k] = __builtin_nontemporal_load(wptr + (size_t)k * PLANE);

  const float* xbase = xin + ((size_t)b * CHN + cc) * PLANE;   // g = 0 plane

  auto issue_tile = [&](int g, int buf) {
    const float* plane = xbase + (size_t)g * WC * PLANE;       // channel g*16+cc
    for (int e = tid; e < HALO_N; e += 256) {
      const int r  = e / HALO_W;
      const int c  = e - r * HALO_W;
      const int ys = y0 + r - PAD3;
      const int xs = x0 + c - PAD3;
      if ((unsigned)ys < (unsigned)IMG && (unsigned)xs < (unsigned)IMG) {
        const unsigned voff = (unsigned)((ys * IMG + xs) * 4);
        const unsigned lb   = (unsigned)(unsigned long long)&tile[buf][e];
        async_f32_to_lds(plane, voff, lb);
      } else {
        tile[buf][e] = 0.0f;                                   // zero halo (disjoint addrs)
      }
    }
  };

  // prologue: fill buffer 0
  issue_tile(0, 0);
  wait_asynccnt0();
  __syncthreads();

  float* obase = out + (((size_t)b * CHN + cc) * IMG + yy) * IMG + xx;

#pragma unroll 1
  for (int g = 0; g < GN; ++g) {
    if (g + 1 < GN) issue_tile(g + 1, (g + 1) & 1);            // prefetch next channel

    const float* lt = &tile[g & 1][0];
    float acc = 0.0f;
#pragma unroll
    for (int kh = 0; kh < KK7; ++kh) {
      const int row = (ty + kh) * HALO_W + tx;
#pragma unroll
      for (int kw = 0; kw < KK7; ++kw)
        acc = __builtin_fmaf(lt[row + kw], w[kh * KK7 + kw], acc);
    }
    __builtin_nontemporal_store(acc, obase + (size_t)g * WC * PLANE);

    if (g + 1 < GN) {
      wait_asynccnt0();      // next buffer landed (this wave's async ops)
      __syncthreads();       // all waves' tiles landed / done reading old buffer
    }
  }
}

extern "C" void kernel_launch(void* const* d_in, const int* in_sizes, int n_in,
                              void* d_out, int out_size, void* d_ws, size_t ws_size,
                              hipStream_t stream) {
  const float* x = (const float*)d_in[0];   // [4,256,64,64]
  const float* w = (const float*)d_in[1];   // [4,1,16,49,64,64]
  float* out = (float*)d_out;               // [4,256,64,64]
  (void)in_sizes; (void)n_in; (void)out_size; (void)d_ws; (void)ws_size;

  dim3 grid((IMG / TW) * (IMG / THT), WC, 4);   // (16, 16, 4)
  dim3 block(256);
  local_conv7x7_async_kernel<<<grid, block, 0, stream>>>(x, w, out);
}